// MultiplicativeAttn_85624468013566
// MI455X (gfx1250) — compile-verified
//
#include <hip/hip_runtime.h>
#include <math.h>
#include <stdint.h>

typedef __attribute__((ext_vector_type(2))) float v2f;
typedef __attribute__((ext_vector_type(8))) float v8f;

#define SEQ 8192
#define HID 4096
#define BLOCK 64        // 2 wave32s
#define ROWS 32         // rows per block (16 per wave)
#define KT 64           // K tile (floats per row per stage)
#define LSTRIDE 68      // LDS row pitch (dwords) = 272B: 16B aligned, conflict-free wmma feed

// Issue one tile (ROWS x KT floats) of async global->LDS copies, 16B per lane.
// gbase points at enc[rowBase*HID]; kf is the K offset of the tile.
__device__ __forceinline__ void issue_tile_async(
    uint32_t lds_base, const float* __restrict__ gbase, int tid, int kf)
{
    #pragma unroll
    for (int j = 0; j < (ROWS * KT) / (4 * BLOCK); ++j) {   // 8 chunks per thread
        const int i  = tid + j * BLOCK;
        const int r  = i >> 4;              // row within tile
        const int kk = (i & 15) << 2;       // float offset within row
        const uint32_t goff = (uint32_t)((r * HID + kf + kk) * 4);      // global byte offset
        const uint32_t lptr = lds_base + (uint32_t)((r * LSTRIDE + kk) * 4); // LDS byte addr
        asm volatile("global_load_async_to_lds_b128 %0, %1, %2"
                     :: "v"(lptr), "v"(goff), "s"(gbase)
                     : "memory");
    }
}

__global__ __launch_bounds__(BLOCK) void attn_scores_kernel(
    const float* __restrict__ hidden,
    const float* __restrict__ enc,
    float* __restrict__ scores)
{
    __shared__ float sh_h[HID];                     // 16 KB: full hidden vector
    __shared__ float sh_a[2][ROWS * LSTRIDE];       // double-buffered encoder tiles

    const int tid  = threadIdx.x;
    const int lane = tid & 31;
    const int wave = tid >> 5;
    const int rowBase = blockIdx.x * ROWS;

    const float* gbase = enc + (size_t)rowBase * HID;
    const uint32_t lds0 = (uint32_t)(uintptr_t)&sh_a[0][0];
    const uint32_t lds1 = (uint32_t)(uintptr_t)&sh_a[1][0];

    // Load full hidden vector into LDS (coalesced float4); published by first barrier.
    {
        const float4* h4  = (const float4*)hidden;
        float4*       sh4 = (float4*)sh_h;
        for (int i = tid; i < HID / 4; i += BLOCK) sh4[i] = h4[i];
    }

    // Prologue: async-fill buffer 0 with the first K tile.
    issue_tile_async(lds0, gbase, tid, 0);

    v8f c = {};
    // WMMA fp32 A-layout: lanes 0-15 -> rows, K={k,k+1}; lanes 16-31 -> same rows, K={k+2,k+3}
    const int myrow = (wave << 4) + (lane & 15);
    const int khalf = (lane >> 4) << 1;             // 0 or 2
    int cur = 0;

    for (int k0 = 0; k0 < HID; k0 += KT) {
        // Our async fills for the current buffer must land, then publish to both waves.
        asm volatile("s_wait_asynccnt 0x0" ::: "memory");
        __syncthreads();

        // Kick off the next tile into the other buffer (overlaps the WMMA loop below).
        if (k0 + KT < HID)
            issue_tile_async(cur ? lds0 : lds1, gbase, tid, k0 + KT);

        const float* a = &sh_a[cur][0];
        #pragma unroll
        for (int kk = 0; kk < KT; kk += 4) {
            const v2f av = *(const v2f*)&a[myrow * LSTRIDE + kk + khalf];  // A 16x4 chunk
            const v2f bv = *(const v2f*)&sh_h[k0 + kk + khalf];            // B: hidden bcast
            c = __builtin_amdgcn_wmma_f32_16x16x4_f32(
                    /*neg_a=*/false, av, /*neg_b=*/false, bv,
                    /*c_mod=*/(short)0, c, /*reuse_a=*/false, /*reuse_b=*/false);
        }
        cur ^= 1;
    }

    // C/D layout: VGPR r, lanes 0-15 -> M=r (N=lane), lanes 16-31 -> M=8+r. Take N=0 columns.
    const int rg = rowBase + (wave << 4);
    if (lane == 0) {
        #pragma unroll
        for (int r = 0; r < 8; ++r) scores[rg + r] = c[r];
    } else if (lane == 16) {
        #pragma unroll
        for (int r = 0; r < 8; ++r) scores[rg + 8 + r] = c[r];
    }
}

__global__ __launch_bounds__(1024) void softmax_kernel(
    const float* __restrict__ scores, float* __restrict__ out)
{
    __shared__ float red[32];
    const int tid  = threadIdx.x;   // 1024 threads = 32 wave32s
    const int lane = tid & 31;
    const int wid  = tid >> 5;

    float v[8];
    float m = -3.402823466e38f;
    #pragma unroll
    for (int j = 0; j < 8; ++j) { v[j] = scores[tid + j * 1024]; m = fmaxf(m, v[j]); }
    #pragma unroll
    for (int off = 16; off > 0; off >>= 1) m = fmaxf(m, __shfl_xor(m, off, 32));
    if (lane == 0) red[wid] = m;
    __syncthreads();
    if (wid == 0) {
        float t = red[lane];
        #pragma unroll
        for (int off = 16; off > 0; off >>= 1) t = fmaxf(t, __shfl_xor(t, off, 32));
        if (lane == 0) red[0] = t;
    }
    __syncthreads();
    m = red[0];
    __syncthreads();

    float s = 0.f;
    #pragma unroll
    for (int j = 0; j < 8; ++j) { v[j] = __expf(v[j] - m); s += v[j]; }
    #pragma unroll
    for (int off = 16; off > 0; off >>= 1) s += __shfl_xor(s, off, 32);
    if (lane == 0) red[wid] = s;
    __syncthreads();
    if (wid == 0) {
        float t = red[lane];
        #pragma unroll
        for (int off = 16; off > 0; off >>= 1) t += __shfl_xor(t, off, 32);
        if (lane == 0) red[0] = t;
    }
    __syncthreads();

    const float inv = 1.f / red[0];
    #pragma unroll
    for (int j = 0; j < 8; ++j) out[tid + j * 1024] = v[j] * inv;
}

extern "C" void kernel_launch(void* const* d_in, const int* in_sizes, int n_in,
                              void* d_out, int out_size, void* d_ws, size_t ws_size,
                              hipStream_t stream) {
    (void)in_sizes; (void)n_in; (void)d_ws; (void)ws_size; (void)out_size;
    const float* hidden = (const float*)d_in[0];   // [1, 4096] fp32
    const float* enc    = (const float*)d_in[1];   // [8192, 1, 4096] fp32
    float*       out    = (float*)d_out;           // [1, 1, 8192] fp32

    attn_scores_kernel<<<SEQ / ROWS, BLOCK, 0, stream>>>(hidden, enc, out);
    softmax_kernel<<<1, 1024, 0, stream>>>(out, out);
}